// Decoder_80874234184280
// MI455X (gfx1250) — compile-verified
//
#include <hip/hip_runtime.h>

typedef __attribute__((ext_vector_type(2))) float v2f;
typedef __attribute__((ext_vector_type(4))) float v4f;
typedef __attribute__((ext_vector_type(8))) float v8f;

#define B_    64
#define C_    23
#define E_    1024
#define D_    1024
#define BOT_  128
#define BINS_ 8192
#define S_    3

// workspace float offsets
#define WE2_OFF 0      // 3*1024 = 3072
#define WB1_OFF 3072   // 3*128  = 384
#define BE2_OFF 3456   // 3
#define PE_OFF  3584   // 64*23*3 = 4416  (total ws use: 8000 floats = 32 KB)

#define TILE_N 64

// ---------------------------------------------------------------------------
// Kernel 1: fold the big weight matrices through the 3-row fc weights.
//   We2[s,e] = sum_d Wfc[s, D+d] * We[d,e]
//   Wb1[s,k] = sum_d Wfc[s, d]   * Wb[d,k]
//   be2[s]   = sum_d Wfc[s, D+d] * be[d]
// ---------------------------------------------------------------------------
__global__ void precompute_kernel(const float* __restrict__ Wb,
                                  const float* __restrict__ We,
                                  const float* __restrict__ be,
                                  const float* __restrict__ Wfc,
                                  float* __restrict__ ws) {
    int gid = blockIdx.x * blockDim.x + threadIdx.x;
    if (gid < 3072) {                       // We2: s = gid/1024, e = gid%1024
        int s = gid >> 10, e = gid & 1023;
        const float* w2 = Wfc + s * 2 * D_ + D_;
        float acc = 0.f;
        for (int d = 0; d < D_; ++d)
            acc = fmaf(w2[d], We[(size_t)d * E_ + e], acc);   // coalesced in e
        ws[WE2_OFF + gid] = acc;
    } else if (gid < 3456) {                // Wb1: s = idx/128, k = idx%128
        int idx = gid - 3072;
        int s = idx >> 7, k = idx & 127;
        const float* w1 = Wfc + s * 2 * D_;
        float acc = 0.f;
        for (int d = 0; d < D_; ++d)
            acc = fmaf(w1[d], Wb[(size_t)d * BOT_ + k], acc); // coalesced in k
        ws[WB1_OFF + idx] = acc;
    } else if (gid < 3459) {                // be2
        int s = gid - 3456;
        const float* w2 = Wfc + s * 2 * D_ + D_;
        float acc = 0.f;
        for (int d = 0; d < D_; ++d)
            acc = fmaf(w2[d], be[d], acc);
        ws[BE2_OFF + s] = acc;
    }
}

// ---------------------------------------------------------------------------
// Kernel 2: pe[b,c,s] = sum_e eos_emb[b,c,e]*We2[s,e] + be2[s] + bfc[s]
// One wave32 per (b,c) pair; lane-strided dot + shuffle reduction.
// ---------------------------------------------------------------------------
__global__ void part_eos_kernel(const float* __restrict__ eos_emb,
                                const float* __restrict__ bfc,
                                float* __restrict__ ws) {
    int wave = (blockIdx.x * blockDim.x + threadIdx.x) >> 5;  // 0..1471
    int lane = threadIdx.x & 31;
    if (wave >= B_ * C_) return;
    const float* row = eos_emb + (size_t)wave * E_;
    const float* we2 = ws + WE2_OFF;
    float a0 = 0.f, a1 = 0.f, a2 = 0.f;
    for (int e = lane; e < E_; e += 32) {
        float x = row[e];
        a0 = fmaf(x, we2[e],          a0);
        a1 = fmaf(x, we2[E_ + e],     a1);
        a2 = fmaf(x, we2[2 * E_ + e], a2);
    }
    for (int off = 16; off > 0; off >>= 1) {
        a0 += __shfl_xor(a0, off, 32);
        a1 += __shfl_xor(a1, off, 32);
        a2 += __shfl_xor(a2, off, 32);
    }
    if (lane == 0) {
        float* pe = ws + PE_OFF + wave * 3;
        pe[0] = a0 + ws[BE2_OFF + 0] + bfc[0];
        pe[1] = a1 + ws[BE2_OFF + 1] + bfc[1];
        pe[2] = a2 + ws[BE2_OFF + 2] + bfc[2];
    }
}

// ---------------------------------------------------------------------------
// Kernel 3: per (c, n-tile of 64):
//   pb[i,s] = sum_k emb_table[1 + c*BINS + n0 + i, k] * Wb1[s,k]   (WMMA f32)
//   out[b,c,n0+i,s] = relu(pb[i,s] + pe[b,c,s])                    (NT stores)
// Block = 128 threads = 4 waves; wave w owns rows [16w, 16w+16).
// ---------------------------------------------------------------------------
__global__ __launch_bounds__(128) void decoder_main_kernel(
        const float* __restrict__ emb_table,
        const float* __restrict__ ws,
        float* __restrict__ out) {
    __shared__ float ELDS[TILE_N * 132];   // emb tile, padded row stride
    __shared__ float BLDS[128 * 16];       // B matrix [k][n], cols>=3 are 0
    __shared__ float PB[TILE_N * 17];      // WMMA result, padded
    __shared__ float PE[B_ * 3];           // part_eos for this c (with biases)

    const int tid = threadIdx.x;
    const int c  = blockIdx.y;
    const int n0 = blockIdx.x * TILE_N;
    const size_t row_base = (size_t)(1 + c * BINS_ + n0) * BOT_;

    // stage 64x128 emb tile, coalesced 128-bit loads, streaming (non-temporal)
    for (int idx = tid; idx < TILE_N * 32; idx += 128) {
        int i = idx >> 5, k4 = idx & 31;
        const v4f* src =
            (const v4f*)(emb_table + row_base + (size_t)i * BOT_) + k4;
        v4f v = __builtin_nontemporal_load(src);
        v4f* dst = (v4f*)&ELDS[i * 132 + k4 * 4];   // 528*i + 16*k4: 16B aligned
        *dst = v;
    }
    // B matrix: BLDS[k*16+n] = (n<3) ? Wb1[n,k] : 0
    for (int idx = tid; idx < 128 * 16; idx += 128) {
        int k = idx >> 4, n = idx & 15;
        BLDS[idx] = (n < S_) ? ws[WB1_OFF + n * BOT_ + k] : 0.f;
    }
    // pe rows for this c
    for (int t = tid; t < B_ * 3; t += 128) {
        int b = t / 3, s = t - b * 3;
        PE[t] = ws[PE_OFF + (b * C_ + c) * 3 + s];
    }
    __syncthreads();

    // WMMA f32 16x16x4, K=128 in 32 steps.
    // A (16x4): lane L holds M=L&15, K pair base = (L>>4)*2  (per ISA layout)
    // B (4x16): lane L holds N=L&15, same K pair striping
    const int wv   = tid >> 5;
    const int lane = tid & 31;
    const int mn   = lane & 15;
    const int kb   = (lane >> 4) * 2;
    const float* arow = &ELDS[(wv * 16 + mn) * 132];
    v8f acc = {};
    for (int kk = 0; kk < 32; ++kk) {
        int k0 = 4 * kk + kb;
        v2f a; a.x = arow[k0];              a.y = arow[k0 + 1];
        v2f b; b.x = BLDS[k0 * 16 + mn];    b.y = BLDS[(k0 + 1) * 16 + mn];
        acc = __builtin_amdgcn_wmma_f32_16x16x4_f32(
                  false, a, false, b, (short)0, acc, false, false);
    }
    {   // D layout: VGPR j, lane L -> row (L>>4)*8 + j, col L&15
        int rbase = wv * 16 + (lane >> 4) * 8;
        #pragma unroll
        for (int j = 0; j < 8; ++j)
            PB[(rbase + j) * 17 + mn] = acc[j];
    }
    __syncthreads();

    // stream the output: for each b, 192 contiguous floats
    for (int idx = tid; idx < B_ * TILE_N * 3; idx += 128) {
        int b = idx / (TILE_N * 3);
        int r = idx - b * (TILE_N * 3);
        int i = r / 3, s = r - i * 3;
        float v = PB[i * 17 + s] + PE[b * 3 + s];
        v = v > 0.f ? v : 0.f;
        size_t o = ((size_t)(b * C_ + c) * BINS_ + (n0 + i)) * 3 + s;
        __builtin_nontemporal_store(v, out + o);
    }
}

// ---------------------------------------------------------------------------
extern "C" void kernel_launch(void* const* d_in, const int* in_sizes, int n_in,
                              void* d_out, int out_size, void* d_ws, size_t ws_size,
                              hipStream_t stream) {
    const float* eos_emb   = (const float*)d_in[0];
    const float* emb_table = (const float*)d_in[1];
    const float* Wb        = (const float*)d_in[2];
    const float* We        = (const float*)d_in[3];
    const float* be        = (const float*)d_in[4];
    const float* Wfc       = (const float*)d_in[5];
    const float* bfc       = (const float*)d_in[6];
    float* out = (float*)d_out;
    float* ws  = (float*)d_ws;

    precompute_kernel<<<14, 256, 0, stream>>>(Wb, We, be, Wfc, ws);
    part_eos_kernel<<<(B_ * C_) / 8, 256, 0, stream>>>(eos_emb, bfc, ws);
    dim3 grid(BINS_ / TILE_N, C_);
    decoder_main_kernel<<<grid, 128, 0, stream>>>(emb_table, ws, out);
}